// Upsample_85779086836269
// MI455X (gfx1250) — compile-verified
//
#include <hip/hip_runtime.h>

// CDNA5 / gfx1250. wave32. f32 WMMA path (V_WMMA_F32_16X16X4_F32) for exact
// f32 Gram matrix; top-k ordering is precision-sensitive so we do NOT downcast.

typedef __attribute__((ext_vector_type(2))) float v2f;
typedef __attribute__((ext_vector_type(8))) float v8f;

#define B_   16
#define C_   64
#define N_   2048
#define K_   8          // neighbors kept (reference: top-9 minus self)
#define WAVES 8
#define CHUNK 64        // columns per inner iteration
#define TILES 4         // CHUNK / 16

// ---------------------------------------------------------------- norms ----
__global__ __launch_bounds__(256) void norms_kernel(const float* __restrict__ x,
                                                    float* __restrict__ xx) {
  int g = blockIdx.x * 256 + threadIdx.x;        // over B*N
  int b = g >> 11;
  int n = g & (N_ - 1);
  const float* xp = x + (size_t)b * C_ * N_ + n;
  float s = 0.f;
#pragma unroll
  for (int c = 0; c < C_; ++c) {
    float v = xp[(size_t)c * N_];
    s += v * v;
  }
  xx[g] = s;
}

// ------------------------------------------------------------- copy half ---
__global__ __launch_bounds__(256) void copy_kernel(const float* __restrict__ x,
                                                   float* __restrict__ out) {
  int g  = blockIdx.x * 256 + threadIdx.x;       // over B*C*N
  int n  = g & (N_ - 1);
  int bc = g >> 11;
  out[(size_t)bc * (2 * N_) + n] = x[g];
}

// ------------------------------------------------------------------ main ---
// grid = B * (N/128); block = 256 (8 waves, 16 rows per wave)
__global__ __launch_bounds__(256) void knn_kernel(const float* __restrict__ x,
                                                  const float* __restrict__ xx,
                                                  float* __restrict__ out) {
  __shared__ float sc[WAVES][16][CHUNK + 1];     // +1 pad: row stride 65 dwords -> no bank conflict
  __shared__ int   il[WAVES][16][K_];

  const int tid  = threadIdx.x;
  const int wave = tid >> 5;
  const int lane = tid & 31;
  const int b      = blockIdx.x >> 4;            // 16 row-blocks per batch
  const int rowBlk = blockIdx.x & 15;
  const int n0     = rowBlk * 128 + wave * 16;   // this wave's 16-row strip
  const size_t xb  = (size_t)b * C_ * N_;

  const int lm = lane & 15;
  const int lh = lane >> 4;                      // half-wave id (0/1)

  // ---- preload A strip (16x64 rows-of-x^T) into registers -----------------
  // f32 WMMA A 16x4 layout: lanes 0-15 hold K=0,1; lanes 16-31 hold K=2,3.
  v2f a[16];
#pragma unroll
  for (int s = 0; s < 16; ++s) {
    int c0 = 4 * s + 2 * lh;
    a[s].x = x[xb + (size_t)c0 * N_ + (n0 + lm)];
    a[s].y = x[xb + (size_t)(c0 + 1) * N_ + (n0 + lm)];
  }

  // per-lane running top-8 (sorted descending), 2 lanes per row
  float tv[K_];
  int   ti[K_];
#pragma unroll
  for (int k = 0; k < K_; ++k) { tv[k] = -3.4e38f; ti[k] = 0; }

  const int srow  = lane >> 1;                   // row this lane scans
  const int shalf = lane & 1;                    // which 32 columns
  const int selfn = n0 + srow;

  for (int chunk = 0; chunk < N_ / CHUNK; ++chunk) {
    const int m0 = chunk * CHUNK;

    v8f acc[TILES] = {};                         // f32 accumulators, zero C

#pragma unroll
    for (int s = 0; s < 16; ++s) {               // K loop: C=64 in steps of 4
      const int c0 = 4 * s + 2 * lh;
      const float* xc0 = x + xb + (size_t)c0 * N_;
      const float* xc1 = xc0 + N_;
      const v2f av = a[s];
#pragma unroll
      for (int t = 0; t < TILES; ++t) {          // 4 column tiles of 16
        const int m = m0 + t * 16 + lm;
        v2f bv;
        bv.x = xc0[m];
        bv.y = xc1[m];
        acc[t] = __builtin_amdgcn_wmma_f32_16x16x4_f32(
            /*neg_a=*/false, av, /*neg_b=*/false, bv,
            /*c_mod=*/(short)0, acc[t], /*reuse_a=*/false, /*reuse_b=*/false);
      }
    }

    // ---- fold -||x_m||^2 in (per-row -||x_n||^2 doesn't change ordering) --
    // D layout: lanes 0-15 vgpr r -> row r; lanes 16-31 vgpr r -> row r+8; col = lane&15
#pragma unroll
    for (int t = 0; t < TILES; ++t) {
      const int m  = m0 + t * 16 + lm;
      const float xm = xx[b * N_ + m];
#pragma unroll
      for (int r = 0; r < 8; ++r) {
        sc[wave][r + 8 * lh][t * 16 + lm] = 2.0f * acc[t][r] - xm;
      }
    }
    asm volatile("s_wait_dscnt 0" ::: "memory"); // intra-wave LDS produce->consume

    // ---- scan: 2 lanes per row, 32 candidates each -------------------------
#pragma unroll 4
    for (int j = 0; j < 32; ++j) {
      const int col = shalf * 32 + j;
      const int m   = m0 + col;
      const float sv = sc[wave][srow][col];
      if (m == selfn) continue;                  // drop self-match (ref drops top-1)
      if (sv > tv[K_ - 1]) {                     // mostly-false fast reject
        float cv = sv; int ci = m;               // carry-insert keeps list sorted
#pragma unroll
        for (int k = 0; k < K_; ++k) {
          const float ov = tv[k]; const int oi = ti[k];
          const bool gt = cv > ov;
          tv[k] = gt ? cv : ov;  ti[k] = gt ? ci : oi;
          cv    = gt ? ov : cv;  ci    = gt ? oi : ci;
        }
      }
    }
    asm volatile("s_wait_dscnt 0" ::: "memory"); // reads done before next-chunk stores
  }

  // ---- merge the two lanes of each row (snapshot partner first) ------------
  float pv[K_]; int pi[K_];
#pragma unroll
  for (int k = 0; k < K_; ++k) {
    pv[k] = __shfl_xor(tv[k], 1, 32);
    pi[k] = __shfl_xor(ti[k], 1, 32);
  }
#pragma unroll
  for (int k = 0; k < K_; ++k) {
    float cv = pv[k]; int ci = pi[k];
    if (cv > tv[K_ - 1]) {
#pragma unroll
      for (int q = 0; q < K_; ++q) {
        const float ov = tv[q]; const int oi = ti[q];
        const bool gt = cv > ov;
        tv[q] = gt ? cv : ov;  ti[q] = gt ? ci : oi;
        cv    = gt ? ov : cv;  ci    = gt ? oi : ci;
      }
    }
  }
  if (shalf == 0) {
#pragma unroll
    for (int k = 0; k < K_; ++k) il[wave][srow][k] = ti[k];
  }
  asm volatile("s_wait_dscnt 0" ::: "memory");

  // ---- neighbor mean: 2 lanes per row, 32 channels each --------------------
  const int row = lane & 15;
  const int n   = n0 + row;
  int nb[K_];
#pragma unroll
  for (int k = 0; k < K_; ++k) nb[k] = il[wave][row][k];

  const int cbase = lh * 32;
  for (int c = 0; c < 32; ++c) {
    const int cc = cbase + c;
    const float* xc = x + xb + (size_t)cc * N_;
    float ssum = 0.f;
#pragma unroll
    for (int k = 0; k < K_; ++k) ssum += xc[nb[k]];
    out[((size_t)b * C_ + cc) * (2 * N_) + N_ + n] = ssum * 0.125f;
  }
}

// --------------------------------------------------------------- launcher --
extern "C" void kernel_launch(void* const* d_in, const int* in_sizes, int n_in,
                              void* d_out, int out_size, void* d_ws, size_t ws_size,
                              hipStream_t stream) {
  const float* x  = (const float*)d_in[0];
  float* out      = (float*)d_out;
  float* xx       = (float*)d_ws;               // B*N floats = 128 KB scratch

  norms_kernel<<<(B_ * N_) / 256, 256, 0, stream>>>(x, xx);
  knn_kernel<<<B_ * (N_ / 128), 256, 0, stream>>>(x, xx, out);
  copy_kernel<<<(B_ * C_ * N_) / 256, 256, 0, stream>>>(x, out);
}